// CDSSM_class_12524124635898
// MI455X (gfx1250) — compile-verified
//
#include <hip/hip_runtime.h>
#include <hip/hip_bf16.h>
#include <math.h>

typedef __attribute__((ext_vector_type(2))) float v2f;
typedef __attribute__((ext_vector_type(8))) float v8f;

#define NPOS   382   // conv output positions per window (384 - 3 + 1)
#define WINF   384   // flat window size (3 tokens * 128 dims)
#define NTILE_N  8   // 128 filters / 16
#define MT_PER_WAVE 3 // 24 position tiles / 8 waves

// One block (8 wave32) per batch item. Block gathers one window into LDS,
// each wave runs 3 M-tiles x 8 N-tiles of V_WMMA_F32_16X16X4_F32.
// The unused K=3 slot carries the bias: patch row = [x[l], x[l+1], x[l+2], 1]
// and B row K=3 = conv_b[f], so the WMMA output is conv+bias directly and the
// epilogue is a pure running max (ReLU = max with 0-initialized accumulator).
__global__ __launch_bounds__(256)
void cdssm_encode_kernel(const int* __restrict__ ids, int T, int nwin,
                         const float* __restrict__ emb,     // (50000,128)
                         const float* __restrict__ conv_w,  // (128,1,3)
                         const float* __restrict__ conv_b,  // (128,)
                         float* __restrict__ pooled)        // (B, 382)
{
    __shared__ float lds[WINF + 4];
    const int b    = blockIdx.x;
    const int tid  = threadIdx.x;
    const int wave = tid >> 5;
    const int lane = tid & 31;
    const int lm   = lane & 15;
    const bool hi  = lane >= 16;

    // B fragments: Wt is (4 x 128) with row 3 = bias.  Layout (symmetric with
    // A 16x4): VGPR r, lanes 0-15 -> K=r, N=lane; lanes 16-31 -> K=r+2.
    v2f bfrag[NTILE_N];
#pragma unroll
    for (int n = 0; n < NTILE_N; ++n) {
        int f = n * 16 + lm;
        float w0 = conv_w[f * 3 + 0];
        float w1 = conv_w[f * 3 + 1];
        float w2 = conv_w[f * 3 + 2];
        float bb = conv_b[f];
        bfrag[n].x = hi ? w2 : w0;   // K=2 : K=0
        bfrag[n].y = hi ? bb : w1;   // K=3 (bias) : K=1
    }

    float run[MT_PER_WAVE][8];
#pragma unroll
    for (int m = 0; m < MT_PER_WAVE; ++m)
#pragma unroll
        for (int r = 0; r < 8; ++r) run[m][r] = 0.0f;   // relu floor

    if (tid < 4) lds[WINF + tid] = 0.0f;                // pad tail for l+2 reads

    for (int win = 0; win < nwin; ++win) {
        __syncthreads();                                // protect prior reads
        {   // cooperative gather: 384 floats, tokens 3*win..3*win+2 (0 if >= T)
            int j = tid;
            int t = win * 3 + (j >> 7);
            float v = 0.0f;
            if (t < T) v = emb[(long)ids[b * T + t] * 128 + (j & 127)];
            lds[j] = v;
            j = tid + 256;
            if (j < WINF) {
                t = win * 3 + (j >> 7);
                v = 0.0f;
                if (t < T) v = emb[(long)ids[b * T + t] * 128 + (j & 127)];
                lds[j] = v;
            }
        }
        __syncthreads();

#pragma unroll
        for (int m = 0; m < MT_PER_WAVE; ++m) {
            // A fragment: patch row l = [x[l], x[l+1], x[l+2], 1.0]
            // lanes 0-15: (K0,K1) = x[l], x[l+1]; lanes 16-31: (K2,K3)=x[l+2],1
            int l0 = (wave * MT_PER_WAVE + m) * 16 + lm;
            v2f a;
            a.x = lds[l0 + (hi ? 2 : 0)];
            a.y = hi ? 1.0f : lds[l0 + 1];
#pragma unroll
            for (int n = 0; n < NTILE_N; ++n) {
                v8f c = {};
                c = __builtin_amdgcn_wmma_f32_16x16x4_f32(
                        false, a, false, bfrag[n], (short)0, c, false, false);
#pragma unroll
                for (int r = 0; r < 8; ++r)
                    run[m][r] = fmaxf(run[m][r], c[r]);  // bias already in c
            }
        }
    }

    // Max over the 16 filter lanes of each N-group, then write valid positions.
#pragma unroll
    for (int m = 0; m < MT_PER_WAVE; ++m) {
#pragma unroll
        for (int r = 0; r < 8; ++r) {
            float v = run[m][r];
            v = fmaxf(v, __shfl_xor(v, 1, 32));
            v = fmaxf(v, __shfl_xor(v, 2, 32));
            v = fmaxf(v, __shfl_xor(v, 4, 32));
            v = fmaxf(v, __shfl_xor(v, 8, 32));
            int l = (wave * MT_PER_WAVE + m) * 16 + r + (hi ? 8 : 0);
            if (lm == 0 && l < NPOS) pooled[b * NPOS + l] = v;
        }
    }
}

// cosine(q,s) -> sigmoid, one block per batch item
__global__ __launch_bounds__(128)
void cdssm_cos_kernel(const float* __restrict__ q, const float* __restrict__ s,
                      float* __restrict__ out)
{
    const int b = blockIdx.x, tid = threadIdx.x;
    float dot = 0.0f, qq = 0.0f, ss = 0.0f;
    for (int l = tid; l < NPOS; l += 128) {
        float qv = q[b * NPOS + l], sv = s[b * NPOS + l];
        dot += qv * sv; qq += qv * qv; ss += sv * sv;
    }
    for (int m = 16; m >= 1; m >>= 1) {
        dot += __shfl_xor(dot, m, 32);
        qq  += __shfl_xor(qq,  m, 32);
        ss  += __shfl_xor(ss,  m, 32);
    }
    __shared__ float sd[4], sq[4], sn[4];
    int wave = tid >> 5, lane = tid & 31;
    if (lane == 0) { sd[wave] = dot; sq[wave] = qq; sn[wave] = ss; }
    __syncthreads();
    if (tid == 0) {
        float D = sd[0] + sd[1] + sd[2] + sd[3];
        float Q = sq[0] + sq[1] + sq[2] + sq[3];
        float S = sn[0] + sn[1] + sn[2] + sn[3];
        float cosv = D / ((sqrtf(Q) + 1e-7f) * (sqrtf(S) + 1e-7f));
        out[b] = 1.0f / (1.0f + expf(-cosv));
    }
}

extern "C" void kernel_launch(void* const* d_in, const int* in_sizes, int n_in,
                              void* d_out, int out_size, void* d_ws, size_t ws_size,
                              hipStream_t stream)
{
    const int*   sentence = (const int*)d_in[0];   // (32, 512)
    const int*   query    = (const int*)d_in[1];   // (32, 16)
    const float* emb      = (const float*)d_in[2]; // (50000, 128)
    const float* conv_w   = (const float*)d_in[3]; // (128, 1, 3)
    const float* conv_b   = (const float*)d_in[4]; // (128,)
    float*       out      = (float*)d_out;         // (32, 1)

    float* pooled_q = (float*)d_ws;                // 32 * 382
    float* pooled_s = pooled_q + 32 * NPOS;        // 32 * 382

    // query:    T=16,  npad=2 -> T'=18,  nwin=6
    cdssm_encode_kernel<<<32, 256, 0, stream>>>(query,    16,  6,   emb, conv_w, conv_b, pooled_q);
    // sentence: T=512, npad=1 -> T'=513, nwin=171
    cdssm_encode_kernel<<<32, 256, 0, stream>>>(sentence, 512, 171, emb, conv_w, conv_b, pooled_s);

    cdssm_cos_kernel<<<32, 128, 0, stream>>>(pooled_q, pooled_s, out);
}